// RDFBins_61787399520645
// MI455X (gfx1250) — compile-verified
//
#include <hip/hip_runtime.h>
#include <hip/hip_bf16.h>
#include <stdint.h>

// ---------------- problem constants (fixed by reference) ----------------
#define SPECIES_N   4
#define BINS_N      100
#define EDGES_N     101
#define HIST_SIZE   (SPECIES_N * SPECIES_N * BINS_N)   // 1600
#define BLOCK       256
#define VEC         4
#define TILE        (BLOCK * VEC)                      // 1024 pairs / tile
#define HIST_GRID   1024
#define INV_W       (100.0f / 6.0f)                    // uniform bin width 0.06

// ---------------- CDNA5 async global->LDS helpers ----------------
typedef int v4i __attribute__((vector_size(16)));
typedef __attribute__((address_space(1))) v4i gm_v4i;
typedef __attribute__((address_space(3))) v4i lds_v4i;

__device__ __forceinline__ void async_ld16(const void* g, void* l) {
#if __has_builtin(__builtin_amdgcn_global_load_async_to_lds_b128)
  __builtin_amdgcn_global_load_async_to_lds_b128(
      (gm_v4i*)(uint64_t)g,
      (lds_v4i*)(uint32_t)(uint64_t)l, 0, 0);
#else
  unsigned loff = (unsigned)(uint64_t)l;   // LDS byte offset = low 32 bits
  asm volatile("global_load_async_to_lds_b128 %0, %1, off"
               :: "v"(loff), "v"(g) : "memory");
#endif
}

__device__ __forceinline__ void wait_async_le3() {
#if __has_builtin(__builtin_amdgcn_s_wait_asynccnt)
  __builtin_amdgcn_s_wait_asynccnt(3);
#else
  asm volatile("s_wait_asynccnt 3" ::: "memory");
#endif
  asm volatile("" ::: "memory");  // compiler barrier: keep LDS reads below
}

__device__ __forceinline__ void wait_async_0() {
#if __has_builtin(__builtin_amdgcn_s_wait_asynccnt)
  __builtin_amdgcn_s_wait_asynccnt(0);
#else
  asm volatile("s_wait_asynccnt 0" ::: "memory");
#endif
  asm volatile("" ::: "memory");
}

// ---------------- kernels ----------------
__global__ void RDF_zero_kernel(unsigned int* __restrict__ p, int n) {
  int i = blockIdx.x * blockDim.x + threadIdx.x;
  if (i < n) p[i] = 0u;
}

// one_hot is strict one-hot bool [n_atoms, 4] (1 byte per bool) -> label 0..3
__global__ void RDF_species_kernel(const unsigned char* __restrict__ onehot,
                                   unsigned char* __restrict__ species,
                                   int nAtoms) {
  int a = blockIdx.x * blockDim.x + threadIdx.x;
  if (a < nAtoms) {
    uchar4 v = ((const uchar4*)onehot)[a];
    species[a] = (unsigned char)(v.y * 1 + v.z * 2 + v.w * 3);
  }
}

__device__ __forceinline__ void rdf_accum(float d, int ia, int ib,
                                          const unsigned char* __restrict__ species,
                                          const float* lbins,
                                          unsigned int* lhist) {
  // searchsorted(bins, d, 'right')-1 for uniform float32 linspace edges:
  // multiply, then exact +-1 fixup against the true float32 edge values.
  if (d < lbins[BINS_N]) {                 // drop d >= bins[-1]
    int b = (int)(d * INV_W);
    b = b > (BINS_N - 1) ? (BINS_N - 1) : b;
    if (d < lbins[b])            b -= 1;
    else if (d >= lbins[b + 1])  b += 1;
    if (b >= 0 && b < BINS_N) {            // drop d < bins[0]
      int si = species[ia];
      int sj = species[ib];
      atomicAdd(&lhist[(si * SPECIES_N + sj) * BINS_N + b], 1u);  // ds_add_u32
    }
  }
}

__global__ __launch_bounds__(BLOCK)
void RDF_hist_kernel(const float* __restrict__ dist,
                     const int* __restrict__ pfirst,
                     const int* __restrict__ psecond,
                     const unsigned char* __restrict__ species,
                     const float* __restrict__ bins,
                     unsigned int* __restrict__ gcount,
                     int fullTiles, int nPairs) {
  __shared__ float4 dbuf[2][BLOCK];
  __shared__ int4   fbuf[2][BLOCK];
  __shared__ int4   sbuf[2][BLOCK];
  __shared__ unsigned int lhist[HIST_SIZE];
  __shared__ float  lbins[EDGES_N];

  const int tid = threadIdx.x;
  for (int i = tid; i < HIST_SIZE; i += BLOCK) lhist[i] = 0u;
  if (tid < EDGES_N) lbins[tid] = bins[tid];
  __syncthreads();

  // ---- double-buffered async-to-LDS pipeline over full tiles ----
  int t  = blockIdx.x;
  int pp = 0;
  if (t < fullTiles) {
    int base = t * TILE + tid * 4;       // each lane async-loads its own 16B
    async_ld16(dist    + base, &dbuf[0][tid]);
    async_ld16(pfirst  + base, &fbuf[0][tid]);
    async_ld16(psecond + base, &sbuf[0][tid]);
  }
  while (t < fullTiles) {
    int tn = t + gridDim.x;
    if (tn < fullTiles) {
      int nb = tn * TILE + tid * 4;
      async_ld16(dist    + nb, &dbuf[pp ^ 1][tid]);
      async_ld16(pfirst  + nb, &fbuf[pp ^ 1][tid]);
      async_ld16(psecond + nb, &sbuf[pp ^ 1][tid]);
      wait_async_le3();                  // current tile's 3 loads complete
    } else {
      wait_async_0();
    }
    // each lane consumes exactly the 16B it async-loaded: no barrier needed
    float4 d4 = dbuf[pp][tid];
    int4   f4 = fbuf[pp][tid];
    int4   s4 = sbuf[pp][tid];
    rdf_accum(d4.x, f4.x, s4.x, species, lbins, lhist);
    rdf_accum(d4.y, f4.y, s4.y, species, lbins, lhist);
    rdf_accum(d4.z, f4.z, s4.z, species, lbins, lhist);
    rdf_accum(d4.w, f4.w, s4.w, species, lbins, lhist);
    pp ^= 1;
    t = tn;
  }

  // ---- tail (< TILE pairs): block 0, plain loads ----
  if (blockIdx.x == 0) {
    for (int p = fullTiles * TILE + tid; p < nPairs; p += BLOCK)
      rdf_accum(dist[p], pfirst[p], psecond[p], species, lbins, lhist);
  }

  __syncthreads();
  for (int i = tid; i < HIST_SIZE; i += BLOCK) {
    unsigned int c = lhist[i];
    if (c) atomicAdd(&gcount[i], c);     // global_atomic_add_u32
  }
}

__global__ void RDF_finalize_kernel(const unsigned int* __restrict__ c,
                                    const int* __restrict__ nmol,
                                    float* __restrict__ out, int n) {
  int i = blockIdx.x * blockDim.x + threadIdx.x;
  if (i < n) out[i] = (float)c[i] * (1.0f / (float)nmol[0]);
}

// ---------------- launcher ----------------
extern "C" void kernel_launch(void* const* d_in, const int* in_sizes, int n_in,
                              void* d_out, int out_size, void* d_ws, size_t ws_size,
                              hipStream_t stream) {
  const float*         dist   = (const float*)d_in[0];
  const int*           pf     = (const int*)d_in[1];
  const int*           ps     = (const int*)d_in[2];
  const unsigned char* onehot = (const unsigned char*)d_in[3];  // bool = 1 byte
  const int*           nmol   = (const int*)d_in[4];
  const float*         bins   = (const float*)d_in[5];

  const int nPairs = in_sizes[0];
  const int nAtoms = in_sizes[3] / SPECIES_N;

  // workspace layout: [0,6400) u32 counts, [6400, 6400+nAtoms) u8 species
  unsigned int*  gcount  = (unsigned int*)d_ws;
  unsigned char* species = (unsigned char*)d_ws + HIST_SIZE * sizeof(unsigned int);

  RDF_zero_kernel<<<(HIST_SIZE + BLOCK - 1) / BLOCK, BLOCK, 0, stream>>>(gcount, HIST_SIZE);
  RDF_species_kernel<<<(nAtoms + BLOCK - 1) / BLOCK, BLOCK, 0, stream>>>(onehot, species, nAtoms);

  int fullTiles = nPairs / TILE;
  int grid = fullTiles < HIST_GRID ? (fullTiles > 0 ? fullTiles : 1) : HIST_GRID;
  RDF_hist_kernel<<<grid, BLOCK, 0, stream>>>(dist, pf, ps, species, bins,
                                              gcount, fullTiles, nPairs);

  RDF_finalize_kernel<<<(HIST_SIZE + BLOCK - 1) / BLOCK, BLOCK, 0, stream>>>(
      gcount, nmol, (float*)d_out, HIST_SIZE);
}